// CSPAN_80582176408287
// MI455X (gfx1250) — compile-verified
//
#include <hip/hip_runtime.h>

// ---------------- CDNA5 WMMA types ----------------
typedef __attribute__((ext_vector_type(2))) float v2f;
typedef __attribute__((ext_vector_type(8))) float v8f;

__device__ __forceinline__ v8f wmma_f32_16x16x4(v2f a, v2f b, v8f c) {
  // D = A(16x4 f32) * B(4x16 f32) + C(16x16 f32), wave32
  return __builtin_amdgcn_wmma_f32_16x16x4_f32(false, a, false, b, (short)0, c,
                                               false, false);
}

// ---------------- problem constants ----------------
#define B_ 8
#define S_ 4096
#define D_ 512
#define M_ 32
#define MD_ 512
#define IMPH_ 256
#define AGGH_ 256
#define WIN_ 64
#define NWIN_ 128
#define NCAT 288 // 32 (motif scores) + 256 (wa1 hidden)

#define INV_SQRT_MD 0.04419417382415922f
#define NEG_INF (-__builtin_inff())

// ---------------- block reductions (blockDim == 256) ----------------
__device__ __forceinline__ float blockReduceSum(float v, float* sred) {
#pragma unroll
  for (int off = 16; off > 0; off >>= 1) v += __shfl_xor(v, off, 32);
  int wid = threadIdx.x >> 5, lane = threadIdx.x & 31;
  if (lane == 0) sred[wid] = v;
  __syncthreads();
  if (threadIdx.x == 0) {
    float r = sred[0];
    for (int i = 1; i < 8; ++i) r += sred[i];
    sred[0] = r;
  }
  __syncthreads();
  float r = sred[0];
  __syncthreads();
  return r;
}

__device__ __forceinline__ float blockReduceMax(float v, float* sred) {
#pragma unroll
  for (int off = 16; off > 0; off >>= 1) v = fmaxf(v, __shfl_xor(v, off, 32));
  int wid = threadIdx.x >> 5, lane = threadIdx.x & 31;
  if (lane == 0) sred[wid] = v;
  __syncthreads();
  if (threadIdx.x == 0) {
    float r = sred[0];
    for (int i = 1; i < 8; ++i) r = fmaxf(r, sred[i]);
    sred[0] = r;
  }
  __syncthreads();
  float r = sred[0];
  __syncthreads();
  return r;
}

// ---------------- prep: pair-packed B operand [256 pairs][288 cols] ----------
// q2p[p][n] = {Bcat[2p][n], Bcat[2p+1][n]}, Bcat[e][m<32]=sum_d fp_w[e][d]*mq[m][d]
//             Bcat[e][32+j]=wa1_w[e][j]
__global__ __launch_bounds__(256) void kP_pack(const float* __restrict__ fp_w,
                                               const float* __restrict__ mq,
                                               const float* __restrict__ wa1_w,
                                               float* __restrict__ q2p) {
  int idx = blockIdx.x * 256 + threadIdx.x;
  if (idx >= 256 * NCAT) return;
  int p = idx / NCAT, n = idx % NCAT;
  v2f v;
  if (n < 32) {
    const float* r0 = fp_w + (size_t)(2 * p) * MD_;
    const float* r1 = fp_w + (size_t)(2 * p + 1) * MD_;
    const float* q = mq + (size_t)n * MD_;
    float a0 = 0.f, a1 = 0.f;
    for (int d = 0; d < MD_; ++d) {
      float qd = q[d];
      a0 += r0[d] * qd;
      a1 += r1[d] * qd;
    }
    v.x = a0;
    v.y = a1;
  } else {
    int j = n - 32;
    v.x = wa1_w[(size_t)(2 * p) * AGGH_ + j];
    v.y = wa1_w[(size_t)(2 * p + 1) * AGGH_ + j];
  }
  ((v2f*)q2p)[idx] = v;
}

// sb[m] = mq[m,:] . fp_b
__global__ void kP_sb(const float* __restrict__ mq,
                      const float* __restrict__ fp_b, float* __restrict__ sb) {
  int m = threadIdx.x;
  if (m < M_) {
    float a = 0.f;
    for (int d = 0; d < MD_; ++d) a += mq[(size_t)m * MD_ + d] * fp_b[d];
    sb[m] = a;
  }
}

// ---------------- Kernel A: fused features x [q2^T | wa1_w] GEMM ------------
// grid (S/16, B), block 256 (8 waves). A tile (16xK=512) staged in LDS.
__global__ __launch_bounds__(256) void kA_gemm(
    const float* __restrict__ feat, const int* __restrict__ mask,
    const float* __restrict__ q2p, const float* __restrict__ sb,
    float* __restrict__ attn, float* __restrict__ h_wa) {
  __shared__ v2f As[16][258]; // 258 pair stride -> 4-bank row skew
  int s0 = blockIdx.x * 16;
  int b = blockIdx.y;
  const float* arow = feat + ((size_t)b * S_ + s0) * D_;
  for (int i = threadIdx.x; i < 16 * 256; i += 256) {
    int r = i >> 8, p = i & 255;
    As[r][p] = *(const v2f*)(arow + (size_t)r * D_ + 2 * p);
  }
  __syncthreads();

  int wave = threadIdx.x >> 5;
  int lane = threadIdx.x & 31;
  int lo = lane & 15, g = lane >> 4;
  const v2f* bmat = (const v2f*)q2p;

  for (int t = wave; t < NCAT / 16; t += 8) {
    int n0 = t * 16;
    v8f acc = {};
#pragma unroll 4
    for (int kp = 0; kp < 256; kp += 2) { // kp = k/2, K step 4
      v2f a = As[lo][kp + g];
      v2f bv = bmat[(size_t)(kp + g) * NCAT + n0 + lo];
      acc = wmma_f32_16x16x4(a, bv, acc);
    }
    if (n0 < 32) { // motif scores -> masked, scaled, into attn region
      int m = n0 + lo;
      float sbv = sb[m];
#pragma unroll
      for (int j = 0; j < 8; ++j) {
        int s = s0 + j + 8 * g;
        float v = (acc[j] + sbv) * INV_SQRT_MD;
        if (mask[b * S_ + s] == 0) v = NEG_INF;
        attn[((size_t)(b * M_ + m)) * S_ + s] = v;
      }
    } else { // wa1 hidden (bias added later in tscore kernel)
      int j2 = (n0 - 32) + lo;
#pragma unroll
      for (int j = 0; j < 8; ++j) {
        int s = s0 + j + 8 * g;
        h_wa[((size_t)(b * S_) + s) * AGGH_ + j2] = acc[j];
      }
    }
  }
}

// ---------------- softmax over s for each (b,m) row -------------------------
__global__ __launch_bounds__(256) void kS_softmax(float* __restrict__ attn) {
  __shared__ float sred[8];
  float* p = attn + (size_t)blockIdx.x * S_;
  float mx = NEG_INF;
  for (int i = threadIdx.x; i < S_; i += 256) mx = fmaxf(mx, p[i]);
  mx = blockReduceMax(mx, sred);
  float sum = 0.f;
  for (int i = threadIdx.x; i < S_; i += 256) {
    float e = expf(p[i] - mx);
    p[i] = e;
    sum += e;
  }
  sum = blockReduceSum(sum, sred);
  float inv = 1.f / sum;
  for (int i = threadIdx.x; i < S_; i += 256) p[i] *= inv;
}

// ---------------- Kernel B: mfeat = attn @ features (per b, K=4096) ---------
// one wave computes both 16-row m-tiles of a 16-col n-tile (B loads shared)
__global__ __launch_bounds__(256) void kB_mfeat(const float* __restrict__ attn,
                                                const float* __restrict__ feat,
                                                float* __restrict__ mfeat) {
  int id = blockIdx.x * 8 + (threadIdx.x >> 5); // 0..255 = b*32 + nt
  int lane = threadIdx.x & 31;
  int lo = lane & 15, g = lane >> 4;
  int b = id >> 5, nt = id & 31;
  int n0 = nt * 16;
  const float* arow0 = attn + ((size_t)(b * M_) + lo) * S_;
  const float* arow1 = attn + ((size_t)(b * M_) + 16 + lo) * S_;
  const float* bcol = feat + (size_t)b * S_ * D_ + n0 + lo;
  v8f acc0 = {}, acc1 = {};
  for (int k = 0; k < S_; k += 4) {
    int kk = k + 2 * g;
    v2f a0 = *(const v2f*)(arow0 + kk);
    v2f a1 = *(const v2f*)(arow1 + kk);
    v2f bv;
    bv.x = bcol[(size_t)kk * D_];
    bv.y = bcol[(size_t)(kk + 1) * D_];
    acc0 = wmma_f32_16x16x4(a0, bv, acc0);
    acc1 = wmma_f32_16x16x4(a1, bv, acc1);
  }
#pragma unroll
  for (int j = 0; j < 8; ++j) {
    int m = j + 8 * g;
    mfeat[((size_t)(b * M_) + m) * D_ + n0 + lo] = acc0[j];
    mfeat[((size_t)(b * M_) + 16 + m) * D_ + n0 + lo] = acc1[j];
  }
}

// ---------------- Kernel C: mf = mfeat @ vp_w + vp_b (256 x 512 x 512) ------
__global__ __launch_bounds__(256) void kC_mf(const float* __restrict__ mfeat,
                                             const float* __restrict__ vp_w,
                                             const float* __restrict__ vp_b,
                                             float* __restrict__ mf) {
  int id = blockIdx.x * 8 + (threadIdx.x >> 5); // 0..511 = mt*32 + nt
  int lane = threadIdx.x & 31;
  int lo = lane & 15, g = lane >> 4;
  int mt = id >> 5, nt = id & 31;
  int m0 = mt * 16, n0 = nt * 16;
  const float* arow = mfeat + (size_t)(m0 + lo) * D_;
  const float* bcol = vp_w + n0 + lo;
  v8f acc = {};
  for (int k = 0; k < D_; k += 4) {
    int kk = k + 2 * g;
    v2f a = *(const v2f*)(arow + kk);
    v2f bv;
    bv.x = bcol[(size_t)kk * MD_];
    bv.y = bcol[(size_t)(kk + 1) * MD_];
    acc = wmma_f32_16x16x4(a, bv, acc);
  }
  float bias = vp_b[n0 + lo];
#pragma unroll
  for (int j = 0; j < 8; ++j) {
    int r = m0 + j + 8 * g;
    mf[(size_t)r * MD_ + n0 + lo] = acc[j] + bias;
  }
}

// ---------------- Kernel D: LayerNorm + importance MLP + sigmoid scale ------
__global__ __launch_bounds__(256) void kD_ln_imp(
    const float* __restrict__ mf, const float* __restrict__ ln_g,
    const float* __restrict__ ln_b, const float* __restrict__ imp1_w,
    const float* __restrict__ imp1_b, const float* __restrict__ imp2_w,
    const float* __restrict__ imp2_b, float* __restrict__ mscaled) {
  __shared__ float smfn[MD_];
  __shared__ float sred[8];
  int row = blockIdx.x; // b*32+m
  const float* x = mf + (size_t)row * MD_;
  int e0 = threadIdx.x, e1 = threadIdx.x + 256;
  float x0 = x[e0], x1 = x[e1];
  float mu = blockReduceSum(x0 + x1, sred) * (1.f / MD_);
  float d0 = x0 - mu, d1 = x1 - mu;
  float var = blockReduceSum(d0 * d0 + d1 * d1, sred) * (1.f / MD_);
  float rstd = rsqrtf(var + 1e-5f);
  float n0v = d0 * rstd * ln_g[e0] + ln_b[e0];
  float n1v = d1 * rstd * ln_g[e1] + ln_b[e1];
  smfn[e0] = n0v;
  smfn[e1] = n1v;
  __syncthreads();
  int j = threadIdx.x;
  float acc = imp1_b[j];
  for (int e = 0; e < MD_; ++e) acc += smfn[e] * imp1_w[(size_t)e * IMPH_ + j];
  float part = fmaxf(acc, 0.f) * imp2_w[j];
  float imp = blockReduceSum(part, sred) + imp2_b[0];
  float iw = 1.f / (1.f + expf(-imp));
  mscaled[(size_t)row * MD_ + e0] = n0v * iw;
  mscaled[(size_t)row * MD_ + e1] = n1v * iw;
}

// ---------------- E1: macc[b,e] = sum_m mscaled -----------------------------
__global__ __launch_bounds__(256) void kE1_macc(
    const float* __restrict__ mscaled, float* __restrict__ macc) {
  int idx = blockIdx.x * 256 + threadIdx.x; // 0..4095
  int b = idx >> 9, e = idx & 511;
  float a = 0.f;
  for (int m = 0; m < M_; ++m) a += mscaled[((size_t)(b * M_ + m)) * MD_ + e];
  macc[idx] = a;
}

// ---------------- E2: motif_out = macc @ mout_w + mout_b --------------------
__global__ __launch_bounds__(256) void kE2_motif(
    const float* __restrict__ macc, const float* __restrict__ mout_w,
    const float* __restrict__ mout_b, float* __restrict__ motif_out) {
  int idx = blockIdx.x * 256 + threadIdx.x; // 0..4095
  int b = idx >> 9, d = idx & 511;
  float a = mout_b[d];
  for (int e = 0; e < MD_; ++e)
    a += macc[b * MD_ + e] * mout_w[(size_t)e * D_ + d];
  motif_out[idx] = a;
}

// ---------------- F: token window scores: tanh(h+b) . wa2 + b2 -------------
__global__ __launch_bounds__(256) void kF_tscore(
    const float* __restrict__ h_wa, const float* __restrict__ wa1_b,
    const float* __restrict__ wa2_w, const float* __restrict__ wa2_b,
    float* __restrict__ tscore) {
  int tok = blockIdx.x * 8 + (threadIdx.x >> 5);
  int lane = threadIdx.x & 31;
  const float* h = h_wa + (size_t)tok * AGGH_;
  float acc = 0.f;
#pragma unroll
  for (int i = 0; i < AGGH_ / 32; ++i) {
    int j = lane + i * 32;
    acc += tanhf(h[j] + wa1_b[j]) * wa2_w[j];
  }
#pragma unroll
  for (int off = 16; off > 0; off >>= 1) acc += __shfl_xor(acc, off, 32);
  if (lane == 0) tscore[tok] = acc + wa2_b[0];
}

// ---------------- G: per-window masked softmax + pooled wf ------------------
__global__ __launch_bounds__(256) void kG_window(const float* __restrict__ feat,
                                                 const int* __restrict__ mask,
                                                 const float* __restrict__ tscore,
                                                 float* __restrict__ wf) {
  __shared__ float sp[WIN_];
  int n = blockIdx.x, b = blockIdx.y;
  int start = n * 32;
  if (start > S_ - WIN_) start = S_ - WIN_; // duplicate final window (ref)
  if (threadIdx.x < WIN_) {
    int s = start + threadIdx.x;
    sp[threadIdx.x] = mask[b * S_ + s] ? tscore[b * S_ + s] : NEG_INF;
  }
  __syncthreads();
  if (threadIdx.x == 0) {
    float mx = NEG_INF;
    for (int w = 0; w < WIN_; ++w) mx = fmaxf(mx, sp[w]);
    float sum = 0.f;
    for (int w = 0; w < WIN_; ++w) {
      float e = expf(sp[w] - mx);
      sp[w] = e;
      sum += e;
    }
    float inv = 1.f / sum;
    for (int w = 0; w < WIN_; ++w) sp[w] *= inv;
  }
  __syncthreads();
  for (int e = threadIdx.x; e < D_; e += 256) {
    const float* fb = feat + ((size_t)b * S_ + start) * D_ + e;
    float acc = 0.f;
#pragma unroll 8
    for (int w = 0; w < WIN_; ++w) acc += sp[w] * fb[(size_t)w * D_];
    wf[((size_t)(b * NWIN_ + n)) * D_ + e] = acc;
  }
}

// ---------------- H: context scores cs[b,n] ---------------------------------
__global__ __launch_bounds__(256) void kH_cs(const float* __restrict__ wf,
                                             const float* __restrict__ ca1_w,
                                             const float* __restrict__ ca1_b,
                                             const float* __restrict__ ca2_w,
                                             const float* __restrict__ ca2_b,
                                             float* __restrict__ cs) {
  __shared__ float sw[D_];
  __shared__ float sred[8];
  int id = blockIdx.x; // b*128+n
  sw[threadIdx.x] = wf[(size_t)id * D_ + threadIdx.x];
  sw[threadIdx.x + 256] = wf[(size_t)id * D_ + threadIdx.x + 256];
  __syncthreads();
  int j = threadIdx.x;
  float acc = ca1_b[j];
  for (int e = 0; e < D_; ++e) acc += sw[e] * ca1_w[(size_t)e * AGGH_ + j];
  float part = fmaxf(acc, 0.f) * ca2_w[j];
  float s = blockReduceSum(part, sred);
  if (threadIdx.x == 0) cs[id] = s + ca2_b[0];
}

// ---------------- I: window softmax + hagg + hier_out -----------------------
__global__ __launch_bounds__(256) void kI_final(const float* __restrict__ wf,
                                                const float* __restrict__ cs,
                                                const float* __restrict__ hout_w,
                                                const float* __restrict__ hout_b,
                                                float* __restrict__ hier) {
  __shared__ float scw[NWIN_];
  __shared__ float shagg[D_];
  int b = blockIdx.x;
  if (threadIdx.x == 0) {
    float mx = NEG_INF;
    for (int n = 0; n < NWIN_; ++n) mx = fmaxf(mx, cs[b * NWIN_ + n]);
    float sum = 0.f;
    for (int n = 0; n < NWIN_; ++n) {
      float e = expf(cs[b * NWIN_ + n] - mx);
      scw[n] = e;
      sum += e;
    }
    float inv = 1.f / sum;
    for (int n = 0; n < NWIN_; ++n) scw[n] *= inv;
  }
  __syncthreads();
  for (int e = threadIdx.x; e < D_; e += 256) {
    float acc = 0.f;
    for (int n = 0; n < NWIN_; ++n)
      acc += scw[n] * wf[((size_t)(b * NWIN_ + n)) * D_ + e];
    shagg[e] = acc;
  }
  __syncthreads();
  for (int d = threadIdx.x; d < D_; d += 256) {
    float acc = hout_b[d];
    for (int e = 0; e < D_; ++e) acc += shagg[e] * hout_w[(size_t)e * D_ + d];
    hier[b * D_ + d] = acc;
  }
}

// ---------------- launch ----------------------------------------------------
extern "C" void kernel_launch(void* const* d_in, const int* in_sizes, int n_in,
                              void* d_out, int out_size, void* d_ws,
                              size_t ws_size, hipStream_t stream) {
  (void)in_sizes;
  (void)n_in;
  (void)out_size;
  (void)ws_size;
  const float* feat = (const float*)d_in[0];
  const int* mask = (const int*)d_in[1];
  const float* mq = (const float*)d_in[2];
  const float* fp_w = (const float*)d_in[3];
  const float* fp_b = (const float*)d_in[4];
  const float* vp_w = (const float*)d_in[5];
  const float* vp_b = (const float*)d_in[6];
  const float* ln_g = (const float*)d_in[7];
  const float* ln_b = (const float*)d_in[8];
  const float* imp1_w = (const float*)d_in[9];
  const float* imp1_b = (const float*)d_in[10];
  const float* imp2_w = (const float*)d_in[11];
  const float* imp2_b = (const float*)d_in[12];
  const float* mout_w = (const float*)d_in[13];
  const float* mout_b = (const float*)d_in[14];
  const float* wa1_w = (const float*)d_in[15];
  const float* wa1_b = (const float*)d_in[16];
  const float* wa2_w = (const float*)d_in[17];
  const float* wa2_b = (const float*)d_in[18];
  const float* ca1_w = (const float*)d_in[19];
  const float* ca1_b = (const float*)d_in[20];
  const float* ca2_w = (const float*)d_in[21];
  const float* ca2_b = (const float*)d_in[22];
  const float* hout_w = (const float*)d_in[23];
  const float* hout_b = (const float*)d_in[24];

  float* out = (float*)d_out;
  float* motif_out = out;                       // [B, D]
  float* attn = out + B_ * D_;                  // [B, M, S]
  float* hier = out + B_ * D_ + B_ * M_ * S_;   // [B, D]

  // workspace layout (floats)
  float* ws = (float*)d_ws;
  float* ws_q2p = ws;                         // 256*288*2 = 147456
  float* ws_sb = ws_q2p + 256 * NCAT * 2;     // 32
  float* ws_hwa = ws_sb + 32;                 // B*S*256 = 8388608
  float* ws_tscore = ws_hwa + (size_t)B_ * S_ * AGGH_;   // 32768
  float* ws_mfeat = ws_tscore + B_ * S_;      // 131072
  float* ws_mf = ws_mfeat + B_ * M_ * D_;     // 131072
  float* ws_mscaled = ws_mf + B_ * M_ * MD_;  // 131072
  float* ws_macc = ws_mscaled + B_ * M_ * MD_; // 4096
  float* ws_wf = ws_macc + B_ * MD_;          // B*128*512 = 524288
  float* ws_cs = ws_wf + (size_t)B_ * NWIN_ * D_; // 1024

  kP_pack<<<(256 * NCAT + 255) / 256, 256, 0, stream>>>(fp_w, mq, wa1_w, ws_q2p);
  kP_sb<<<1, 32, 0, stream>>>(mq, fp_b, ws_sb);
  kA_gemm<<<dim3(S_ / 16, B_), 256, 0, stream>>>(feat, mask, ws_q2p, ws_sb,
                                                 attn, ws_hwa);
  kS_softmax<<<B_ * M_, 256, 0, stream>>>(attn);
  kB_mfeat<<<B_ * M_ / 8, 256, 0, stream>>>(attn, feat, ws_mfeat);
  kC_mf<<<64, 256, 0, stream>>>(ws_mfeat, vp_w, vp_b, ws_mf);
  kD_ln_imp<<<B_ * M_, 256, 0, stream>>>(ws_mf, ln_g, ln_b, imp1_w, imp1_b,
                                         imp2_w, imp2_b, ws_mscaled);
  kE1_macc<<<B_ * MD_ / 256, 256, 0, stream>>>(ws_mscaled, ws_macc);
  kE2_motif<<<B_ * D_ / 256, 256, 0, stream>>>(ws_macc, mout_w, mout_b,
                                               motif_out);
  kF_tscore<<<B_ * S_ / 8, 256, 0, stream>>>(ws_hwa, wa1_b, wa2_w, wa2_b,
                                             ws_tscore);
  kG_window<<<dim3(NWIN_, B_), 256, 0, stream>>>(feat, mask, ws_tscore, ws_wf);
  kH_cs<<<B_ * NWIN_, 256, 0, stream>>>(ws_wf, ca1_w, ca1_b, ca2_w, ca2_b,
                                        ws_cs);
  kI_final<<<B_, 256, 0, stream>>>(ws_wf, ws_cs, hout_w, hout_b, hier);
}